// NeRFPointSampler_58961311040064
// MI455X (gfx1250) — compile-verified
//
#include <hip/hip_runtime.h>
#include <stdint.h>

// NeRF importance sampler for MI455X (gfx1250).
// One ray per wave32; 8 independent waves (256 threads) per block.
// HBM-bound (~0.95 GB traffic, ~40us roofline at 23.3 TB/s): all bulk data
// movement goes through the CDNA5 async global<->LDS data movers (ASYNCcnt);
// CDF built with a wave32 shfl scan; sampling+merge are LDS binary searches.
// All sharing is intra-wave -> no block barriers, only wave-local fences +
// explicit s_wait_asynccnt / s_wait_dscnt cross-pipeline waits.

#define RAYS_PER_BLOCK 8
#define THREADS 256
#define FLOATS_PER_RAY 1152  // 64 z + 64 w + 64 bins + 64 cdf + 128 smp + 192 merged + 576 pts
#define EPS_W 1e-5f

// CDNA5: low 32 bits of a flat shared-aperture address == LDS byte offset
// (CDNA5 ISA), which is what the async ops' LDS operand wants.
__device__ __forceinline__ uint32_t lds_off(const void* p) {
  return (uint32_t)(uintptr_t)p;
}

__device__ __forceinline__ void async_load_b64(void* lds_dst, const void* gsrc) {
  asm volatile("global_load_async_to_lds_b64 %0, %1, off"
               :: "v"(lds_off(lds_dst)), "v"(gsrc) : "memory");
}
__device__ __forceinline__ void async_store_b64(const void* lds_src, void* gdst) {
  asm volatile("global_store_async_from_lds_b64 %1, %0, off"
               :: "v"(lds_off(lds_src)), "v"(gdst) : "memory");
}
__device__ __forceinline__ void async_store_b128(const void* lds_src, void* gdst) {
  asm volatile("global_store_async_from_lds_b128 %1, %0, off"
               :: "v"(lds_off(lds_src)), "v"(gdst) : "memory");
}
__device__ __forceinline__ void wait_async0() { asm volatile("s_wait_asynccnt 0" ::: "memory"); }
__device__ __forceinline__ void wait_ds0()    { asm volatile("s_wait_dscnt 0"    ::: "memory"); }

// Intra-wave "barrier": CDNA5 LDS ops from one wave execute in order, so we
// only need to stop the compiler from reordering LDS accesses across stages.
__device__ __forceinline__ void wave_fence() {
  __builtin_amdgcn_wave_barrier();
  asm volatile("" ::: "memory");
}

__global__ __launch_bounds__(THREADS)
void nerf_point_sampler_kernel(const float* __restrict__ rays_o,
                               const float* __restrict__ rays_d,
                               const float* __restrict__ z_vals,
                               const float* __restrict__ weights,
                               float* __restrict__ pts,      // [N,192,3]
                               float* __restrict__ z_out,    // [N,192]
                               int n_rays) {
  __shared__ __align__(16) float smem[RAYS_PER_BLOCK * FLOATS_PER_RAY];  // 36 KB

  const int lane = threadIdx.x & 31;
  const int wave = threadIdx.x >> 5;
  const int ray  = blockIdx.x * RAYS_PER_BLOCK + wave;
  if (ray >= n_rays) return;  // uniform per wave; N divides evenly anyway

  float* sz  = &smem[wave * FLOATS_PER_RAY]; // z_vals row      [64]
  float* sw  = sz + 64;                      // weights row     [64]
  float* sb  = sz + 128;                     // bins            [63]
  float* sc  = sz + 192;                     // cdf             [63]
  float* ss  = sz + 256;                     // fine samples    [128]
  float* szs = sz + 384;                     // merged z        [192]
  float* sp  = sz + 576;                     // staged pts      [576]

  // ---- Stage 0: async DMA both rows into LDS; overlap o/d loads under it ----
  const size_t rbase = (size_t)ray * 64 + 2 * lane;
  async_load_b64(&sz[2 * lane], z_vals  + rbase);
  async_load_b64(&sw[2 * lane], weights + rbase);

  const float ox = rays_o[3 * (size_t)ray + 0];
  const float oy = rays_o[3 * (size_t)ray + 1];
  const float oz = rays_o[3 * (size_t)ray + 2];
  const float dx = rays_d[3 * (size_t)ray + 0];
  const float dy = rays_d[3 * (size_t)ray + 1];
  const float dz = rays_d[3 * (size_t)ray + 2];

  wait_async0();
  wave_fence();

  // ---- Stage 1: bins + normalized CDF (wave32 shfl scan, 2 elems/lane) ----
  for (int i = lane; i < 63; i += 32)
    sb[i] = 0.5f * (sz[i] + sz[i + 1]);

  {
    // weights_[k] = w[k+1] + eps, k = 0..61; lane l (<31) owns k = 2l, 2l+1
    float a = 0.f, b = 0.f;
    if (lane < 31) {
      a = sw[2 * lane + 1] + EPS_W;
      b = sw[2 * lane + 2] + EPS_W;
    }
    const float pair = a + b;
    float incl = pair;
#pragma unroll
    for (int off = 1; off < 32; off <<= 1) {
      const float n = __shfl_up(incl, off, 32);
      if (lane >= off) incl += n;
    }
    const float total = __shfl(incl, 31, 32);
    const float inv   = 1.0f / total;
    const float excl  = incl - pair;
    if (lane == 0) sc[0] = 0.f;
    if (lane < 31) {
      sc[2 * lane + 1] = (excl + a) * inv;
      sc[2 * lane + 2] = (excl + a + b) * inv;
    }
  }
  wave_fence();

  // ---- Stage 2: inverse-CDF sampling, 4 deterministic u per lane ----
#pragma unroll
  for (int j = 0; j < 4; ++j) {
    const int ui  = lane + 32 * j;
    const float u = (float)ui * (1.0f / 127.0f);   // linspace(0,1,128)
    int lo = 0, hi = 63;                           // searchsorted(cdf, u, 'right')
    while (lo < hi) {
      const int mid = (lo + hi) >> 1;
      if (sc[mid] <= u) lo = mid + 1; else hi = mid;
    }
    int below = lo - 1; if (below < 0)  below = 0;
    int above = lo;     if (above > 62) above = 62;
    const float cb = sc[below], ca = sc[above];
    const float bb = sb[below], ba = sb[above];
    float denom = ca - cb;
    if (denom < 1e-5f) denom = 1.f;
    const float t = (u - cb) / denom;
    ss[ui] = fmaf(t, ba - bb, bb);
  }
  wave_fence();

  // ---- Stage 3: merge two sorted runs (64 z + 128 samples) by rank ----
#pragma unroll
  for (int j = 0; j < 2; ++j) {
    const int i   = lane + 32 * j;
    const float v = sz[i];
    int lo = 0, hi = 128;                          // #samples strictly < v
    while (lo < hi) {
      const int mid = (lo + hi) >> 1;
      if (ss[mid] < v) lo = mid + 1; else hi = mid;
    }
    szs[i + lo] = v;
  }
#pragma unroll
  for (int j = 0; j < 4; ++j) {
    const int i   = lane + 32 * j;
    const float v = ss[i];
    int lo = 0, hi = 64;                           // #z <= v
    while (lo < hi) {
      const int mid = (lo + hi) >> 1;
      if (sz[mid] <= v) lo = mid + 1; else hi = mid;
    }
    szs[i + lo] = v;
  }
  wave_fence();
  wait_ds0();  // merged scatter complete before the async engine reads LDS

  // ---- Stage 4a: z_samples row leaves via the async data mover ----
  {
    float* zo = z_out + (size_t)ray * 192;
    async_store_b128(&szs[4 * lane], zo + 4 * lane);             // 128 floats
    async_store_b64(&szs[128 + 2 * lane], zo + 128 + 2 * lane);  // + 64 floats
  }

  // ---- Stage 4b: pts: each lane computes a contiguous 18-float segment
  // (18 % 3 == 0 -> static x/y/z pattern, no per-element selects), stages it
  // to LDS as b64 writes, then the 2304B row leaves via async b128 stores.
  {
    const int base = lane * 18;
#pragma unroll
    for (int t = 0; t < 3; ++t) {
      const float za = szs[lane * 6 + 2 * t + 0];
      const float zb = szs[lane * 6 + 2 * t + 1];
      const float a0 = fmaf(dx, za, ox);
      const float a1 = fmaf(dy, za, oy);
      const float a2 = fmaf(dz, za, oz);
      const float b0 = fmaf(dx, zb, ox);
      const float b1 = fmaf(dy, zb, oy);
      const float b2 = fmaf(dz, zb, oz);
      float2* d2 = (float2*)&sp[base + 6 * t];   // 8B-aligned (base even)
      d2[0] = make_float2(a0, a1);
      d2[1] = make_float2(a2, b0);
      d2[2] = make_float2(b1, b2);
    }
  }
  wave_fence();
  wait_ds0();  // pts staging complete before the async engine reads LDS
  {
    float* po = pts + (size_t)ray * 576;
#pragma unroll
    for (int j = 0; j < 4; ++j)
      async_store_b128(&sp[128 * j + 4 * lane], po + 128 * j + 4 * lane);  // 512 floats
    async_store_b64(&sp[512 + 2 * lane], po + 512 + 2 * lane);             // + 64 floats
  }

  wait_async0();  // drain all async stores (s_endpgm would also wait-idle)
}

extern "C" void kernel_launch(void* const* d_in, const int* in_sizes, int n_in,
                              void* d_out, int out_size, void* d_ws, size_t ws_size,
                              hipStream_t stream) {
  const float* rays_o  = (const float*)d_in[0];
  const float* rays_d  = (const float*)d_in[1];
  const float* z_vals  = (const float*)d_in[2];
  const float* weights = (const float*)d_in[3];

  const int n_rays = in_sizes[0] / 3;            // rays_o is [N,3]
  float* pts   = (float*)d_out;                  // [N,192,3] first
  float* z_out = pts + (size_t)n_rays * 192 * 3; // then [N,192]

  const int blocks = (n_rays + RAYS_PER_BLOCK - 1) / RAYS_PER_BLOCK;
  nerf_point_sampler_kernel<<<blocks, THREADS, 0, stream>>>(
      rays_o, rays_d, z_vals, weights, pts, z_out, n_rays);
}